// LearnedChannelSelfAttentionPooling_V4_43052752175581
// MI455X (gfx1250) — compile-verified
//
#include <hip/hip_runtime.h>
#include <hip/hip_bf16.h>

typedef float v2f __attribute__((ext_vector_type(2)));
typedef float v4f __attribute__((ext_vector_type(4)));
typedef float v8f __attribute__((ext_vector_type(8)));
typedef unsigned int u32x4 __attribute__((ext_vector_type(4)));
typedef int i32x4 __attribute__((ext_vector_type(4)));
typedef int i32x8 __attribute__((ext_vector_type(8)));

// Problem constants (B=1, C=72, H=W=64)
constexpr int NCH  = 18;    // chunks
constexpr int CS   = 4;     // channels per chunk
constexpr int CTOT = 72;    // total channels
constexpr int NPOS = 4096;  // H*W
constexpr int NTIL = 256;   // NPOS / 16
// Fold 1/sqrt(N) AND log2(e) into q: scores come out pre-scaled for exp2,
// so the hot loops use raw v_exp_f32 with no prescale mul.
constexpr float QSCALE = 1.4426950408889634f / 64.0f;

// native v_exp_f32 (base-2 exponential)
static __device__ __forceinline__ float fast_exp2(float x) {
  return __builtin_amdgcn_exp2f(x);
}

// Layouts:
//   chunk q/k/v : [n][pos][4]   (channel-interleaved; one b64/b128 per operand)
//   global q/k/v: [pos][72]     (channel-interleaved)
//   att         : [n][c][pos]   (row-major, written once)
//   og          : [72][pos]     (row-major, written once)

// Workspace layout (in floats)
constexpr size_t SZ_CH   = (size_t)NCH * CS * NPOS;        // 294912 (== CTOT*NPOS)
constexpr size_t OFF_QC  = 0;
constexpr size_t OFF_KC  = OFF_QC + SZ_CH;
constexpr size_t OFF_VC  = OFF_KC + SZ_CH;
constexpr size_t OFF_QG  = OFF_VC + SZ_CH;
constexpr size_t OFF_KG  = OFF_QG + SZ_CH;
constexpr size_t OFF_VG  = OFF_KG + SZ_CH;
constexpr size_t OFF_ZC  = OFF_VG + SZ_CH;                 // NCH*NPOS
constexpr size_t OFF_ZG  = OFF_ZC + (size_t)NCH * NPOS;    // 4*NPOS (split partials)
constexpr size_t OFF_ATT = OFF_ZG + 4 * (size_t)NPOS;
constexpr size_t OFF_OG  = OFF_ATT + SZ_CH;

static __device__ __forceinline__ v8f wmma_16x16x4_f32(v2f a, v2f b, v8f c) {
  // D = A(16x4,f32) * B(4x16,f32) + C(16x16,f32)
  return __builtin_amdgcn_wmma_f32_16x16x4_f32(false, a, false, b, (short)0, c,
                                               false, false);
}

static __device__ __forceinline__ float reduce16(float v) {
  // sum over the 16 lanes of each wave-half (lanes 0-15 / 16-31)
  v += __shfl_xor(v, 1, 32);
  v += __shfl_xor(v, 2, 32);
  v += __shfl_xor(v, 4, 32);
  v += __shfl_xor(v, 8, 32);
  return v;
}

// ---------------------------------------------------------------------------
// TDM: stage a contiguous run of `nelem` fp32 from global into LDS.
// Descriptor per CDNA5 ISA (2-D, data_size=4B,
// tile = nelem x 1, groups 2/3 zero).
static __device__ __forceinline__ void tdm_load_f32(const float* gsrc,
                                                    unsigned lds_off,
                                                    unsigned nelem) {
  unsigned long long ga = (unsigned long long)(uintptr_t)gsrc;
  u32x4 g0;
  g0[0] = 1u;                                              // count=1, no gather
  g0[1] = lds_off;                                         // LDS byte address
  g0[2] = (unsigned)ga;                                    // global addr [31:0]
  g0[3] = (unsigned)((ga >> 32) & 0x1FFFFFFu) | (2u << 30);// addr[56:32] | type=2
  i32x8 g1;
  g1[0] = (int)(2u << 16);                                 // data_size = 4 bytes
  g1[1] = (int)((nelem & 0xFFFFu) << 16);                  // tensor_dim0 [15:0]
  g1[2] = (int)((nelem >> 16) | (1u << 16));               // dim0 hi | tensor_dim1=1
  g1[3] = (int)((nelem & 0xFFFFu) << 16);                  // tile_dim0 = nelem
  g1[4] = 1;                                               // tile_dim1 = 1
  g1[5] = (int)nelem;                                      // tensor_dim0_stride
  g1[6] = 0;
  g1[7] = 0;
  i32x4 z4 = {0, 0, 0, 0};
#if __clang_major__ >= 23
  i32x8 z8 = {0, 0, 0, 0, 0, 0, 0, 0};
  __builtin_amdgcn_tensor_load_to_lds(g0, g1, z4, z4, z8, 0);
#else
  __builtin_amdgcn_tensor_load_to_lds(g0, g1, z4, z4, 0);
#endif
}

// ---------------------------------------------------------------------------
// Kernel 1: per-chunk QKV (4x4 channel mix). q stores QSCALE folded in.
// Outputs channel-interleaved [n][p][4] -> one b128 store per array.
__global__ void k_qkv_chunk(const float* __restrict__ x,
                            const float* __restrict__ Wq, const float* __restrict__ bq,
                            const float* __restrict__ Wk, const float* __restrict__ bk,
                            const float* __restrict__ Wv, const float* __restrict__ bv,
                            float* __restrict__ q, float* __restrict__ k,
                            float* __restrict__ v) {
  int t = blockIdx.x * blockDim.x + threadIdx.x;   // t < 18*4096
  int n = t >> 12, p = t & 4095;
  float xin[CS];
#pragma unroll
  for (int c = 0; c < CS; ++c) xin[c] = x[(size_t)(n * CS + c) * NPOS + p];
  const float* wq = Wq + n * 16; const float* wk = Wk + n * 16; const float* wv = Wv + n * 16;
  v4f oq, ok, ov;
#pragma unroll
  for (int cc = 0; cc < CS; ++cc) {
    float aq = bq[n * CS + cc], ak = bk[n * CS + cc], av = bv[n * CS + cc];
#pragma unroll
    for (int c = 0; c < CS; ++c) {
      aq += wq[cc * CS + c] * xin[c];
      ak += wk[cc * CS + c] * xin[c];
      av += wv[cc * CS + c] * xin[c];
    }
    oq[cc] = aq * QSCALE; ok[cc] = ak; ov[cc] = av;
  }
  size_t o = (size_t)t * CS;
  *(v4f*)&q[o] = oq; *(v4f*)&k[o] = ok; *(v4f*)&v[o] = ov;
}

// Kernel 2: global QKV (72x72 channel mix), channel-interleaved [p][72].
// One thread per (pos, 8-channel block): two aligned b128 stores per array.
__global__ void k_qkv_global(const float* __restrict__ x,
                             const float* __restrict__ Wq, const float* __restrict__ bq,
                             const float* __restrict__ Wk, const float* __restrict__ bk,
                             const float* __restrict__ Wv, const float* __restrict__ bv,
                             float* __restrict__ q, float* __restrict__ k,
                             float* __restrict__ v) {
  int t = blockIdx.x * blockDim.x + threadIdx.x;   // t < 4096*9
  int cb = t % 9, p = t / 9;
  float aq[8], ak[8], av[8];
#pragma unroll
  for (int u = 0; u < 8; ++u) {
    aq[u] = bq[cb * 8 + u]; ak[u] = bk[cb * 8 + u]; av[u] = bv[cb * 8 + u];
  }
  for (int ci = 0; ci < CTOT; ++ci) {
    float xv = x[(size_t)ci * NPOS + p];
#pragma unroll
    for (int u = 0; u < 8; ++u) {
      int co = cb * 8 + u;
      aq[u] += Wq[co * CTOT + ci] * xv;
      ak[u] += Wk[co * CTOT + ci] * xv;
      av[u] += Wv[co * CTOT + ci] * xv;
    }
  }
  size_t o = (size_t)p * CTOT + cb * 8;
  v4f q0, q1, k0, k1, v0, v1;
#pragma unroll
  for (int u = 0; u < 4; ++u) {
    q0[u] = aq[u] * QSCALE; q1[u] = aq[u + 4] * QSCALE;
    k0[u] = ak[u];          k1[u] = ak[u + 4];
    v0[u] = av[u];          v1[u] = av[u + 4];
  }
  *(v4f*)&q[o] = q0; *(v4f*)&q[o + 4] = q1;
  *(v4f*)&k[o] = k0; *(v4f*)&k[o + 4] = k1;
  *(v4f*)&v[o] = v0; *(v4f*)&v[o + 4] = v1;
}

// ---------------------------------------------------------------------------
// Kernel 3: per-chunk Z[j] = sum_i exp(s[j,i]).  Block = 8 waves sharing
// chunk n; k[n] panel (64KB) staged once via TDM.  exp2-domain scores; four
// independent accumulators break the serial add chain.
__global__ void k_z_chunk(const float* __restrict__ q, const float* __restrict__ k,
                          float* __restrict__ Z) {
  __shared__ float lk[CS * NPOS];                  // 64 KB, [pos][4]
  int n  = blockIdx.x >> 5;                        // 18 chunks * 32 blocks
  int jt = ((blockIdx.x & 31) << 3) + (threadIdx.x >> 5);
  int lane = threadIdx.x & 31;
  int hi = lane >> 4, l = lane & 15, c0 = hi * 2;
  if (threadIdx.x < 32) {                          // wave 0 issues the TDM copy
    tdm_load_f32(k + (size_t)n * (CS * NPOS), (unsigned)(uintptr_t)lk, CS * NPOS);
    __builtin_amdgcn_s_wait_tensorcnt(0);
  }
  __syncthreads();
  const float* qn = q + (size_t)n * (CS * NPOS);
  int j = jt * 16 + l;
  v2f b = *(const v2f*)&qn[j * CS + c0];
  float z0 = 0.f, z1 = 0.f, z2 = 0.f, z3 = 0.f;
  for (int it = 0; it < NTIL; ++it) {
    int i = it * 16 + l;
    v2f a = *(const v2f*)&lk[i * CS + c0];
    v8f acc = {};
    acc = wmma_16x16x4_f32(a, b, acc);
    z0 += fast_exp2(acc[0]); z1 += fast_exp2(acc[1]);
    z2 += fast_exp2(acc[2]); z3 += fast_exp2(acc[3]);
    z0 += fast_exp2(acc[4]); z1 += fast_exp2(acc[5]);
    z2 += fast_exp2(acc[6]); z3 += fast_exp2(acc[7]);
  }
  float zacc = (z0 + z1) + (z2 + z3);
  zacc += __shfl_xor(zacc, 16, 32);
  if (lane < 16) Z[(size_t)n * NPOS + jt * 16 + lane] = zacc;
}

// Kernel 4: global Z (K=72 = 18 accumulating WMMA steps); i-range split by 4.
__global__ void k_z_global(const float* __restrict__ q, const float* __restrict__ k,
                           float* __restrict__ Zp) {
  int wid  = blockIdx.x * 8 + (threadIdx.x >> 5);  // < 256*4
  int lane = threadIdx.x & 31;
  int split = wid & 3, jt = wid >> 2;
  int hi = lane >> 4, l = lane & 15, c0 = hi * 2;
  int j = jt * 16 + l;
  v2f bq[18];
#pragma unroll
  for (int t = 0; t < 18; ++t)
    bq[t] = *(const v2f*)&q[(size_t)j * CTOT + 4 * t + c0];
  float z0 = 0.f, z1 = 0.f, z2 = 0.f, z3 = 0.f;
  for (int it = split * 64; it < split * 64 + 64; ++it) {
    int i = it * 16 + l;
    v8f acc = {};
#pragma unroll
    for (int t = 0; t < 18; ++t) {
      v2f a = *(const v2f*)&k[(size_t)i * CTOT + 4 * t + c0];
      acc = wmma_16x16x4_f32(a, bq[t], acc);
    }
    z0 += fast_exp2(acc[0]); z1 += fast_exp2(acc[1]);
    z2 += fast_exp2(acc[2]); z3 += fast_exp2(acc[3]);
    z0 += fast_exp2(acc[4]); z1 += fast_exp2(acc[5]);
    z2 += fast_exp2(acc[6]); z3 += fast_exp2(acc[7]);
  }
  float zacc = (z0 + z1) + (z2 + z3);
  zacc += __shfl_xor(zacc, 16, 32);
  if (lane < 16) Zp[(size_t)split * NPOS + jt * 16 + lane] = zacc;
}

// Kernel 5: fold 1/Z into v (chunk), in place.  b128 load/store per thread.
__global__ void k_vtilde_chunk(float* __restrict__ v, const float* __restrict__ Z) {
  int t = blockIdx.x * blockDim.x + threadIdx.x;   // < 18*4096
  float zr = 1.0f / Z[t];
  v4f vv = *(v4f*)&v[(size_t)t * CS];
#pragma unroll
  for (int c = 0; c < CS; ++c) vv[c] *= zr;
  *(v4f*)&v[(size_t)t * CS] = vv;
}

// Kernel 5b: fold 1/Z into vG (sums the 4 split partials), in place.
__global__ void k_vtilde_global(float* __restrict__ v, const float* __restrict__ Zp) {
  int t = blockIdx.x * blockDim.x + threadIdx.x;   // < 72*4096
  int j = t / CTOT;                                // [p][72] layout
  float z = Zp[j] + Zp[NPOS + j] + Zp[2 * NPOS + j] + Zp[3 * NPOS + j];
  v[t] *= (1.0f / z);
}

// ---------------------------------------------------------------------------
// Kernel 6: attended[c,i] = sum_j vt[c,j] * exp(T[i,j]).  Block = 8 waves
// sharing chunk n; vt[n] panel (64KB) staged via TDM; per j-tile: one b64 B
// load + one conflict-free ds_load_b128 for the 4 V-weights.
__global__ void k_att_chunk(const float* __restrict__ q, const float* __restrict__ k,
                            const float* __restrict__ vt, float* __restrict__ att) {
  __shared__ float lv[CS * NPOS];                  // 64 KB, [pos][4]
  int n  = blockIdx.x >> 5;
  int it = ((blockIdx.x & 31) << 3) + (threadIdx.x >> 5);
  int lane = threadIdx.x & 31;
  int hi = lane >> 4, l = lane & 15, c0 = hi * 2;
  if (threadIdx.x < 32) {
    tdm_load_f32(vt + (size_t)n * (CS * NPOS), (unsigned)(uintptr_t)lv, CS * NPOS);
    __builtin_amdgcn_s_wait_tensorcnt(0);
  }
  __syncthreads();
  const float* qn = q + (size_t)n * (CS * NPOS);
  const float* kn = k + (size_t)n * (CS * NPOS);
  int i = it * 16 + l;
  v2f a = *(const v2f*)&kn[i * CS + c0];
  float s[CS][8] = {};
  for (int jt = 0; jt < NTIL; ++jt) {
    int j = jt * 16 + l;
    v2f b = *(const v2f*)&qn[j * CS + c0];
    v4f w = *(const v4f*)&lv[j * CS];
    v8f acc = {};
    acc = wmma_16x16x4_f32(a, b, acc);
#pragma unroll
    for (int r = 0; r < 8; ++r) {
      float e = fast_exp2(acc[r]);
      s[0][r] += w[0] * e; s[1][r] += w[1] * e;
      s[2][r] += w[2] * e; s[3][r] += w[3] * e;
    }
  }
#pragma unroll
  for (int c = 0; c < CS; ++c)
#pragma unroll
    for (int r = 0; r < 8; ++r) {
      float val = reduce16(s[c][r]);
      if (l == 0)
        att[(size_t)n * (CS * NPOS) + (size_t)c * NPOS + it * 16 + hi * 8 + r] = val;
    }
}

// Kernel 7: global attended, split over 9 blocks of 8 output channels.
// 18 b64 B loads + two b128 V loads per j-tile.
__global__ void k_att_global(const float* __restrict__ q, const float* __restrict__ k,
                             const float* __restrict__ vt, float* __restrict__ og) {
  int wid  = blockIdx.x * 8 + (threadIdx.x >> 5);  // < 256*9
  int lane = threadIdx.x & 31;
  int it = wid / 9, cb = wid % 9;
  int hi = lane >> 4, l = lane & 15, c0 = hi * 2;
  int i = it * 16 + l;
  v2f ak[18];
#pragma unroll
  for (int t = 0; t < 18; ++t)
    ak[t] = *(const v2f*)&k[(size_t)i * CTOT + 4 * t + c0];
  float s[8][8] = {};
  for (int jt = 0; jt < NTIL; ++jt) {
    int j = jt * 16 + l;
    v8f acc = {};
#pragma unroll
    for (int t = 0; t < 18; ++t) {
      v2f b = *(const v2f*)&q[(size_t)j * CTOT + 4 * t + c0];
      acc = wmma_16x16x4_f32(ak[t], b, acc);
    }
    v4f w0 = *(const v4f*)&vt[(size_t)j * CTOT + cb * 8];
    v4f w1 = *(const v4f*)&vt[(size_t)j * CTOT + cb * 8 + 4];
#pragma unroll
    for (int r = 0; r < 8; ++r) {
      float e = fast_exp2(acc[r]);
#pragma unroll
      for (int cc = 0; cc < 4; ++cc) {
        s[cc][r]     += w0[cc] * e;
        s[cc + 4][r] += w1[cc] * e;
      }
    }
  }
#pragma unroll
  for (int cc = 0; cc < 8; ++cc)
#pragma unroll
    for (int r = 0; r < 8; ++r) {
      float val = reduce16(s[cc][r]);
      if (l == 0)
        og[(size_t)(cb * 8 + cc) * NPOS + it * 16 + hi * 8 + r] = val;
    }
}

// Kernel 8: pooled[n,p] = sum_c xc[n,c,p] * (att + outG)[n,c,p]
__global__ void k_combine(const float* __restrict__ x, const float* __restrict__ att,
                          const float* __restrict__ og, float* __restrict__ out) {
  int t = blockIdx.x * blockDim.x + threadIdx.x;   // < 18*4096
  int n = t >> 12, p = t & 4095;
  float acc = 0.f;
#pragma unroll
  for (int c = 0; c < CS; ++c) {
    size_t gch = (size_t)(n * CS + c) * NPOS + p;
    float comb = att[(size_t)n * (CS * NPOS) + (size_t)c * NPOS + p] + og[gch];
    acc += x[gch] * comb;
  }
  out[t] = acc;
}

// ---------------------------------------------------------------------------
extern "C" void kernel_launch(void* const* d_in, const int* in_sizes, int n_in,
                              void* d_out, int out_size, void* d_ws, size_t ws_size,
                              hipStream_t stream) {
  (void)in_sizes; (void)n_in; (void)out_size; (void)ws_size;
  const float* x   = (const float*)d_in[0];
  const float* Wq  = (const float*)d_in[1];
  const float* bq  = (const float*)d_in[2];
  const float* Wk  = (const float*)d_in[3];
  const float* bk  = (const float*)d_in[4];
  const float* Wv  = (const float*)d_in[5];
  const float* bv  = (const float*)d_in[6];
  const float* WqG = (const float*)d_in[7];
  const float* bqG = (const float*)d_in[8];
  const float* WkG = (const float*)d_in[9];
  const float* bkG = (const float*)d_in[10];
  const float* WvG = (const float*)d_in[11];
  const float* bvG = (const float*)d_in[12];
  float* out = (float*)d_out;
  float* ws  = (float*)d_ws;

  float* qc = ws + OFF_QC; float* kc = ws + OFF_KC; float* vc = ws + OFF_VC;
  float* qg = ws + OFF_QG; float* kg = ws + OFF_KG; float* vg = ws + OFF_VG;
  float* zc = ws + OFF_ZC; float* zg = ws + OFF_ZG;
  float* at = ws + OFF_ATT; float* og = ws + OFF_OG;

  k_qkv_chunk <<<288, 256, 0, stream>>>(x, Wq, bq, Wk, bk, Wv, bv, qc, kc, vc);
  k_qkv_global<<<144, 256, 0, stream>>>(x, WqG, bqG, WkG, bkG, WvG, bvG, qg, kg, vg);
  k_z_chunk   <<<576, 256, 0, stream>>>(qc, kc, zc);        // 18*32 blocks, TDM staged
  k_z_global  <<<128, 256, 0, stream>>>(qg, kg, zg);        // 256*4 waves
  k_vtilde_chunk <<<288, 256, 0, stream>>>(vc, zc);
  k_vtilde_global<<<1152, 256, 0, stream>>>(vg, zg);
  k_att_chunk <<<576, 256, 0, stream>>>(qc, kc, vc, at);    // 18*32 blocks, TDM staged
  k_att_global<<<288, 256, 0, stream>>>(qg, kg, vg, og);    // 256*9 waves
  k_combine   <<<288, 256, 0, stream>>>(x, at, og, out);
}